// Conv2d_30073361006680
// MI455X (gfx1250) — compile-verified
//
#include <hip/hip_runtime.h>

typedef __attribute__((ext_vector_type(2))) float v2f;
typedef __attribute__((ext_vector_type(8))) float v8f;

#define IN_CH   128
#define OUT_CH  256
#define H_IN    56
#define W_IN    56
#define OH      54
#define OW      54
#define NBATCH  32
#define HWIN    (H_IN * W_IN)        // 3136
#define HWOUT   (OH * OW)            // 2916
#define KGEMM   (IN_CH * 9)          // 1152
#define MGEMM   (NBATCH * HWOUT)     // 93312

#define BM 128
#define BN 128
#define BK 16
#define LDT 18                       // padded LDS row stride (floats)
#define NKB (KGEMM / BK)             // 72 K-blocks

#if __has_builtin(__builtin_amdgcn_global_load_async_to_lds_b32)
#define USE_ASYNC 1
#else
#define USE_ASYNC 0
#endif

#if USE_ASYNC
typedef __attribute__((address_space(1))) int gas_int;
typedef __attribute__((address_space(3))) int lds_int;
static __device__ __forceinline__ lds_int* to_lds(float* p) {
    // generic shared address: low 32 bits are the LDS offset (ISA aperture rule)
    return (lds_int*)(unsigned int)(unsigned long long)p;
}
static __device__ __forceinline__ gas_int* to_gas(const float* p) {
    return (gas_int*)(unsigned long long)p;
}
#endif

// K-order: k' = (kh*3+kw)*128 + ic  ->  within a BK=16 block, (kh,kw) is uniform
// and ic is contiguous, so all 8 per-thread gathers are immediate offsets.
static __device__ __forceinline__
void stage_tile(const float* __restrict__ xg, const float* __restrict__ wg,
                float* __restrict__ As_w, float* __restrict__ Bs_w, int it)
{
    const int r   = it >> 3;          // 0..8 == kh*3+kw (uniform, SALU)
    const int kh  = r / 3;
    const int kw  = r - kh * 3;
    const int ic0 = (it & 7) * 16;    // input-channel base of this K block
    const float* pA = xg + ic0 * HWIN + kh * W_IN + kw;
    const float* pB = wg + ic0 * 9 + r;
#if USE_ASYNC
#pragma unroll
    for (int j = 0; j < 8; ++j)
        __builtin_amdgcn_global_load_async_to_lds_b32(to_gas(pA + j * HWIN),
                                                      to_lds(As_w + j), 0, 0);
#pragma unroll
    for (int j = 0; j < 8; ++j)
        __builtin_amdgcn_global_load_async_to_lds_b32(to_gas(pB + j * 9),
                                                      to_lds(Bs_w + j), 0, 0);
#else
#pragma unroll
    for (int j = 0; j < 8; ++j) As_w[j] = pA[j * HWIN];
#pragma unroll
    for (int j = 0; j < 8; ++j) Bs_w[j] = pB[j * 9];
#endif
}

static __device__ __forceinline__ void wait_stage()
{
#if USE_ASYNC
    asm volatile("s_wait_asynccnt 0" ::: "memory");
#endif
}

#define KSTEP_BLOCK(ASR0, ASR1, BSR)                                          \
    _Pragma("unroll")                                                         \
    for (int ks = 0; ks < BK; ks += 4) {                                      \
        v2f a0 = *(const v2f*)((ASR0) + ks);                                  \
        v2f a1 = *(const v2f*)((ASR1) + ks);                                  \
        _Pragma("unroll")                                                     \
        for (int t = 0; t < 4; ++t) {                                         \
            v2f b = *(const v2f*)((BSR) + t * 16 * LDT + ks);                 \
            acc0[t] = __builtin_amdgcn_wmma_f32_16x16x4_f32(                  \
                false, a0, false, b, (short)0, acc0[t], false, false);        \
            acc1[t] = __builtin_amdgcn_wmma_f32_16x16x4_f32(                  \
                false, a1, false, b, (short)0, acc1[t], false, false);        \
        }                                                                     \
    }

__launch_bounds__(256, 2)
__global__ void conv3x3_igemm_wmma(const float* __restrict__ x,
                                   const float* __restrict__ w,
                                   const float* __restrict__ bias,
                                   float* __restrict__ out)
{
    __shared__ float As0[BM * LDT], As1[BM * LDT];   // A: [m_local][k_local]
    __shared__ float Bs0[BN * LDT], Bs1[BN * LDT];   // B: [oc_local][k_local]

    const int tid  = threadIdx.x;
    const int lane = tid & 31;
    const int wave = tid >> 5;
    const int wm   = wave & 3;     // wave M group: rows [wm*32, wm*32+32)
    const int wn   = wave >> 2;    // wave N group: cols [wn*64, wn*64+64)

    const int block_m = blockIdx.x;   // 0..728
    const int block_n = blockIdx.y;   // 0..1

    // ---- staging mapping: thread fills row (tid&127), k (tid>>7)*8 + j ----
    const int s_row = tid & 127;
    const int s_k0  = (tid >> 7) * 8;

    const int m_ld  = block_m * BM + s_row;
    const int n_img = m_ld / HWOUT;
    const int hw_ld = m_ld - n_img * HWOUT;
    const int oh_ld = hw_ld / OW;
    const int ow_ld = hw_ld - oh_ld * OW;
    const float* xg = x + (size_t)n_img * (IN_CH * HWIN)
                        + oh_ld * W_IN + ow_ld + s_k0 * HWIN;
    const float* wg = w + (size_t)(block_n * BN + s_row) * KGEMM + s_k0 * 9;

    float* As_w0 = &As0[s_row * LDT + s_k0];
    float* As_w1 = &As1[s_row * LDT + s_k0];
    float* Bs_w0 = &Bs0[s_row * LDT + s_k0];
    float* Bs_w1 = &Bs1[s_row * LDT + s_k0];

    // ---- fragment geometry (wave32 f32 WMMA layouts) ----
    const int colN    = lane & 15;          // M row (A) / N col (B,C,D) of lane
    const int frag_kp = (lane >> 4) * 2;    // upper half-lanes hold K=2,3

    v8f acc0[4], acc1[4];                   // 32M x 64N per wave, bias-seeded
    {
        const float* bsrc = bias + block_n * BN + wn * 64 + colN;
#pragma unroll
        for (int t = 0; t < 4; ++t) {
            float bv = bsrc[t * 16];
            acc0[t] = (v8f){bv, bv, bv, bv, bv, bv, bv, bv};
            acc1[t] = acc0[t];
        }
    }

    const int ar0 = (wm * 32 +  0 + colN) * LDT + frag_kp;
    const int ar1 = (wm * 32 + 16 + colN) * LDT + frag_kp;
    const int br  = (wn * 64 + colN) * LDT + frag_kp;
    const float* As0_r0 = &As0[ar0]; const float* As0_r1 = &As0[ar1];
    const float* As1_r0 = &As1[ar0]; const float* As1_r1 = &As1[ar1];
    const float* Bs0_r  = &Bs0[br];
    const float* Bs1_r  = &Bs1[br];

    // ---- software-pipelined main loop (double-buffered LDS) ----
    stage_tile(xg, wg, As_w0, Bs_w0, 0);

    for (int it = 0; it < NKB; it += 2) {
        wait_stage();
        __syncthreads();                       // buf0 ready; buf1 free
        if (it + 1 < NKB) stage_tile(xg, wg, As_w1, Bs_w1, it + 1);
        KSTEP_BLOCK(As0_r0, As0_r1, Bs0_r);    // 32 WMMAs overlap staging

        wait_stage();
        __syncthreads();                       // buf1 ready; buf0 free
        if (it + 2 < NKB) stage_tile(xg, wg, As_w0, Bs_w0, it + 2);
        KSTEP_BLOCK(As1_r0, As1_r1, Bs1_r);
    }

    // ---- writeback: D layout -> VGPR v, lane l: M = v + 8*(l>>4), N = l&15 ----
    const int m_half = (lane >> 4) * 8;
#pragma unroll
    for (int s = 0; s < 2; ++s) {
        const v8f* accp = s ? acc1 : acc0;
#pragma unroll
        for (int v = 0; v < 8; ++v) {
            const int m  = block_m * BM + wm * 32 + s * 16 + m_half + v;
            const int ni = m / HWOUT;
            const int hw = m - ni * HWOUT;
            float* orow = out + (size_t)ni * (OUT_CH * HWOUT) + hw
                              + (size_t)(block_n * BN + wn * 64 + colN) * HWOUT;
#pragma unroll
            for (int t = 0; t < 4; ++t)
                orow[(size_t)t * 16 * HWOUT] = accp[t][v];
        }
    }
}

extern "C" void kernel_launch(void* const* d_in, const int* in_sizes, int n_in,
                              void* d_out, int out_size, void* d_ws, size_t ws_size,
                              hipStream_t stream) {
    const float* x    = (const float*)d_in[0];
    const float* w    = (const float*)d_in[1];
    const float* bias = (const float*)d_in[2];
    float* out        = (float*)d_out;

    dim3 grid(MGEMM / BM, OUT_CH / BN, 1);   // 729 x 2
    dim3 block(256, 1, 1);                    // 8 waves (wave32)
    conv3x3_igemm_wmma<<<grid, block, 0, stream>>>(x, w, bias, out);
}